// NodeBlock_17703855194352
// MI455X (gfx1250) — compile-verified
//
#include <hip/hip_runtime.h>
#include <math.h>

#define FEAT 128
#define INV_SQRT_F 0.08838834764831843f   // 1/sqrt(128)

typedef float v2f __attribute__((ext_vector_type(2)));
typedef float v8f __attribute__((ext_vector_type(8)));

// float atomic-max via int/uint punning (valid with -inf init, handles sign mix)
__device__ __forceinline__ void atomic_max_f32(float* addr, float val) {
  if (val >= 0.0f) atomicMax((int*)addr, __float_as_int(val));
  else             atomicMin((unsigned int*)addr, __float_as_uint(val));
}

__global__ void init_ws_kernel(float* __restrict__ agg, float* __restrict__ segmax,
                               float* __restrict__ denom, int N) {
  int t = blockIdx.x * blockDim.x + threadIdx.x;
  if (t < N * FEAT) agg[t] = 0.0f;
  if (t < N) { segmax[t] = -INFINITY; denom[t] = 0.0f; }
}

// One wave32 per edge-direction: 128-float dot product + scatter max
__global__ void __launch_bounds__(256)
edge_logits_kernel(const float* __restrict__ node_attr,
                   const float* __restrict__ edge_attr,
                   const int*   __restrict__ edge_index,
                   float* __restrict__ logits,
                   float* __restrict__ segmax,
                   int E) {
  const int wave = (int)((blockIdx.x * blockDim.x + threadIdx.x) >> 5);
  const int lane = threadIdx.x & 31;
  if (wave >= 2 * E) return;

  int idx;
  const float* ef;
  if (wave < E) {  // first half: edge_attr[:, 0:F], dst = receivers = edge_index[1]
    idx = edge_index[E + wave];
    ef  = edge_attr + (size_t)wave * (2 * FEAT);
  } else {         // second half: edge_attr[:, F:2F], dst = senders = edge_index[0]
    idx = edge_index[wave - E];
    ef  = edge_attr + (size_t)(wave - E) * (2 * FEAT) + FEAT;
  }

  float4 a = *(const float4*)(ef + lane * 4);
  float4 x = ((const float4*)(node_attr + (size_t)idx * FEAT))[lane];
  float s = a.x * x.x + a.y * x.y + a.z * x.z + a.w * x.w;
  #pragma unroll
  for (int off = 16; off >= 1; off >>= 1) s += __shfl_xor(s, off, 32);
  s *= INV_SQRT_F;
  if (lane == 0) {
    logits[wave] = s;
    atomic_max_f32(&segmax[idx], s);
  }
}

// Second edge pass: exp-weight and scatter-add (un-normalized; /denom folded later)
__global__ void __launch_bounds__(256)
edge_agg_kernel(const float* __restrict__ edge_attr,
                const int*   __restrict__ edge_index,
                const float* __restrict__ logits,
                const float* __restrict__ segmax,
                float* __restrict__ denom,
                float* __restrict__ agg,
                int E) {
  const int wave = (int)((blockIdx.x * blockDim.x + threadIdx.x) >> 5);
  const int lane = threadIdx.x & 31;
  if (wave >= 2 * E) return;

  int idx;
  const float* ef;
  if (wave < E) {
    idx = edge_index[E + wave];
    ef  = edge_attr + (size_t)wave * (2 * FEAT);
  } else {
    idx = edge_index[wave - E];
    ef  = edge_attr + (size_t)(wave - E) * (2 * FEAT) + FEAT;
  }

  const float ex = expf(logits[wave] - segmax[idx]);
  if (lane == 0) atomicAdd(&denom[idx], ex);

  float4 a = *(const float4*)(ef + lane * 4);
  float* ag = agg + (size_t)idx * FEAT + lane * 4;
  atomicAdd(ag + 0, a.x * ex);
  atomicAdd(ag + 1, a.y * ex);
  atomicAdd(ag + 2, a.z * ex);
  atomicAdd(ag + 3, a.w * ex);
}

// agg[i,:] *= (denom[i] > 0 ? 1/denom[i] : 0), in place, float4-wide
__global__ void normalize_kernel(float* __restrict__ agg,
                                 const float* __restrict__ denom, int N) {
  int t = blockIdx.x * blockDim.x + threadIdx.x;
  if (t >= N * (FEAT / 4)) return;
  int node = t >> 5;  // FEAT/4 == 32
  float d = denom[node];
  float r = (d > 0.0f) ? 1.0f / d : 0.0f;
  float4* p = (float4*)agg + t;
  float4 v = *p;
  v.x *= r; v.y *= r; v.z *= r; v.w *= r;
  *p = v;
}

// out[N,128] = concat(anorm, node_attr)[N,256] @ W[256,128] + b
// One wave per 16-row strip, 8 column tiles of 16, fp32 WMMA 16x16x4.
__global__ void __launch_bounds__(256)
gemm_wmma_kernel(const float* __restrict__ anorm,
                 const float* __restrict__ node_attr,
                 const float* __restrict__ W,
                 const float* __restrict__ bias,
                 float* __restrict__ out, int N) {
  const int lane  = threadIdx.x & 31;
  const int wv    = threadIdx.x >> 5;
  const int strip = blockIdx.x * 8 + wv;
  const int row0  = strip * 16;
  if (row0 >= N) return;

  const int r    = lane & 15;   // A: row within tile / B,D: column within tile
  const int half = lane >> 4;   // selects K pair (A/B) and row half (D)
  const int rowA = min(row0 + r, N - 1);

  v8f acc[8];
  #pragma unroll
  for (int t = 0; t < 8; ++t) { v8f z = {}; acc[t] = z; }

  for (int k0 = 0; k0 < 2 * FEAT; k0 += 4) {
    const int kb = k0 + half * 2;  // this lane's K columns: kb, kb+1
    const float* srcA = (kb < FEAT)
        ? (anorm     + (size_t)rowA * FEAT + kb)
        : (node_attr + (size_t)rowA * FEAT + (kb - FEAT));
    v2f afrag; afrag.x = srcA[0]; afrag.y = srcA[1];

    const float* w0 = W + (size_t)kb * FEAT;  // rows kb, kb+1 of W
    #pragma unroll
    for (int t = 0; t < 8; ++t) {
      const int n = t * 16 + r;
      v2f bfrag; bfrag.x = w0[n]; bfrag.y = w0[FEAT + n];
      acc[t] = __builtin_amdgcn_wmma_f32_16x16x4_f32(
          false, afrag, false, bfrag, (short)0, acc[t], false, false);
    }
  }

  // Epilogue: single bounds test per wave; unguarded fast path for full strips
  // (D layout: VGPR v holds row (half*8 + v), column n = t*16 + r.)
  if (row0 + 16 <= N) {
    #pragma unroll
    for (int t = 0; t < 8; ++t) {
      const int n = t * 16 + r;
      const float bv = bias[n];
      float* o = out + (size_t)(row0 + half * 8) * FEAT + n;
      #pragma unroll
      for (int v = 0; v < 8; ++v) {
        o[(size_t)v * FEAT] = acc[t][v] + bv;
      }
    }
  } else {
    #pragma unroll
    for (int t = 0; t < 8; ++t) {
      const int n = t * 16 + r;
      const float bv = bias[n];
      #pragma unroll
      for (int v = 0; v < 8; ++v) {
        const int m = row0 + half * 8 + v;
        if (m < N) out[(size_t)m * FEAT + n] = acc[t][v] + bv;
      }
    }
  }
}

extern "C" void kernel_launch(void* const* d_in, const int* in_sizes, int n_in,
                              void* d_out, int out_size, void* d_ws, size_t ws_size,
                              hipStream_t stream) {
  const float* node_attr  = (const float*)d_in[0];
  const float* edge_attr  = (const float*)d_in[1];
  const int*   edge_index = (const int*)d_in[2];
  const float* W          = (const float*)d_in[3];
  const float* b          = (const float*)d_in[4];
  float* out = (float*)d_out;

  const int N = in_sizes[0] / FEAT;
  const int E = in_sizes[1] / (2 * FEAT);

  // workspace layout (all 16B-aligned): agg[N*F] | segmax[N] | denom[N] | logits[2E]
  float* agg    = (float*)d_ws;
  float* segmax = agg + (size_t)N * FEAT;
  float* denom  = segmax + N;
  float* logits = denom + N;

  {
    int total = N * FEAT;
    init_ws_kernel<<<(total + 255) / 256, 256, 0, stream>>>(agg, segmax, denom, N);
  }
  {
    int blocks = (2 * E + 7) / 8;  // 8 waves (one edge-direction each) per block
    edge_logits_kernel<<<blocks, 256, 0, stream>>>(node_attr, edge_attr, edge_index,
                                                   logits, segmax, E);
    edge_agg_kernel<<<blocks, 256, 0, stream>>>(edge_attr, edge_index, logits,
                                                segmax, denom, agg, E);
  }
  {
    int total = N * (FEAT / 4);
    normalize_kernel<<<(total + 255) / 256, 256, 0, stream>>>(agg, denom, N);
  }
  {
    int strips = (N + 15) / 16;
    int blocks = (strips + 7) / 8;
    gemm_wmma_kernel<<<blocks, 256, 0, stream>>>(agg, node_attr, W, b, out, N);
  }
}